// NCutLossOptimized_74560632259477
// MI455X (gfx1250) — compile-verified
//
#include <hip/hip_runtime.h>
#include <math.h>

// Problem constants (match reference)
#define HH   224
#define WWD  224
#define NB   8
#define NK   4
#define RAD  5
#define KSZ  11          // 2*RAD+1
#define TILE 16
#define HALO (TILE + 2*RAD)   // 26

typedef __attribute__((ext_vector_type(2))) float v2f;
typedef __attribute__((ext_vector_type(8))) float v8f;

// CDNA5 async global->LDS copy path (ASYNCcnt-tracked, no VGPR round-trip).
// Builtin prototype (from compiler diagnostics): (int AS1*, int AS3*, imm, imm)
#if defined(__gfx1250__) && __has_builtin(__builtin_amdgcn_global_load_async_to_lds_b32)
#define USE_ASYNC_LDS 1
typedef __attribute__((address_space(1))) int* gas_t;
typedef __attribute__((address_space(3))) int* las_t;
#define ASYNC_CP_B32(g, l) \
    __builtin_amdgcn_global_load_async_to_lds_b32((gas_t)(g), (las_t)(l), 0, 0)
#else
#define USE_ASYNC_LDS 0
#endif

static __device__ __forceinline__ void wait_async_zero() {
#if defined(__gfx1250__)
#if __has_builtin(__builtin_amdgcn_s_wait_asynccnt)
    __builtin_amdgcn_s_wait_asynccnt(0);
#else
    asm volatile("s_wait_asynccnt 0" ::: "memory");
#endif
#endif
}

// ---------------------------------------------------------------------------
// Stage 0: zero the 64 global accumulators (harness does not re-zero ws)
// ---------------------------------------------------------------------------
__global__ __launch_bounds__(64) void ncut_zero(float* __restrict__ acc) {
    acc[threadIdx.x] = 0.0f;
}

// ---------------------------------------------------------------------------
// Stage 1: fused bilateral-weight + per-pixel num/den + block reduction.
// One block = one 16x16 pixel tile of one batch image. 256 threads = 8 waves.
// ---------------------------------------------------------------------------
__global__ __launch_bounds__(256) void ncut_stage1(const float* __restrict__ img,
                                                   const float* __restrict__ lab,
                                                   float* __restrict__ acc)
{
    __shared__ float  s_img[HALO][HALO];     // raw intensity, zero-padded
    __shared__ float4 s_lab[HALO][HALO];     // 4 class probs packed -> b128 LDS ops
    __shared__ float  s_dist[KSZ * KSZ];     // spatial gaussian table
    __shared__ float  s_red[8][256];         // 8 reduction quantities per thread

    const int tid = threadIdx.x;
    const int b   = blockIdx.z;
    const int gx0 = blockIdx.x * TILE - RAD;
    const int gy0 = blockIdx.y * TILE - RAD;

    const float* imgb = img + (size_t)b * (HH * WWD);
    const float* labb = lab + (size_t)b * (NK * HH * WWD);

    // --- cooperative tile+halo load (zero padding outside the image) ---
    for (int i = tid; i < HALO * HALO; i += 256) {
        const int ly = i / HALO, lx = i % HALO;
        const int y = gy0 + ly, x = gx0 + lx;
        const bool in = ((unsigned)y < (unsigned)HH) && ((unsigned)x < (unsigned)WWD);
#if USE_ASYNC_LDS
        if (in) {
            const int o = y * WWD + x;
            float* lslot = (float*)&s_lab[ly][lx];
            ASYNC_CP_B32(imgb + o,                 &s_img[ly][lx]);
            ASYNC_CP_B32(labb + 0 * HH * WWD + o,  lslot + 0);
            ASYNC_CP_B32(labb + 1 * HH * WWD + o,  lslot + 1);
            ASYNC_CP_B32(labb + 2 * HH * WWD + o,  lslot + 2);
            ASYNC_CP_B32(labb + 3 * HH * WWD + o,  lslot + 3);
        } else {
            // plain DS stores; disjoint addresses from the async-written ones
            s_img[ly][lx] = 0.0f;
            s_lab[ly][lx] = make_float4(0.0f, 0.0f, 0.0f, 0.0f);
        }
#else
        float  iv = 0.0f;
        float4 lv = make_float4(0.0f, 0.0f, 0.0f, 0.0f);
        if (in) {
            const int o = y * WWD + x;
            iv   = imgb[o];
            lv.x = labb[0 * HH * WWD + o];
            lv.y = labb[1 * HH * WWD + o];
            lv.z = labb[2 * HH * WWD + o];
            lv.w = labb[3 * HH * WWD + o];
        }
        s_img[ly][lx] = iv;
        s_lab[ly][lx] = lv;
#endif
    }
    // spatial distance table: exp(-(dy^2+dx^2)/sigma_x^2) -- overlaps the copy
    for (int i = tid; i < KSZ * KSZ; i += 256) {
        const float dy = (float)(i / KSZ - RAD);
        const float dx = (float)(i % KSZ - RAD);
        s_dist[i] = __expf(-(dy * dy + dx * dx) * (1.0f / 16.0f));
    }
#if USE_ASYNC_LDS
    wait_async_zero();   // this wave's async copies landed; barrier covers the rest
#endif
    __syncthreads();

    // --- per-pixel bilateral accumulation ---
    // LDS holds raw intensities; reference scales by 255 before differencing:
    // (255*(r-c))^2 / 100 = (r-c)^2 * 650.25
    const int tx = tid & (TILE - 1);
    const int ty = tid >> 4;
    const int gx = blockIdx.x * TILE + tx;
    const int gy = blockIdx.y * TILE + ty;

    const float cI = s_img[ty + RAD][tx + RAD];
    float  den = 0.0f;
    float4 num = make_float4(0.0f, 0.0f, 0.0f, 0.0f);

    for (int dy = 0; dy < KSZ; ++dy) {
        const int  yy = gy + dy - RAD;
        const bool vy = (unsigned)yy < (unsigned)HH;
        #pragma unroll
        for (int dx = 0; dx < KSZ; ++dx) {
            const int  xx = gx + dx - RAD;
            const bool v  = vy && ((unsigned)xx < (unsigned)WWD);
            const float r = s_img[ty + dy][tx + dx];
            const float d = r - cI;
            float w = s_dist[dy * KSZ + dx] * __expf(-d * d * 650.25f);
            w = v ? w : 0.0f;                // boundary mask (matches unfold zero-pad)
            const float4 lv = s_lab[ty + dy][tx + dx];
            den  += w;
            num.x = fmaf(w, lv.x, num.x);
            num.y = fmaf(w, lv.y, num.y);
            num.z = fmaf(w, lv.z, num.z);
            num.w = fmaf(w, lv.w, num.w);
        }
    }

    // per-pixel contributions to the 8 global sums:
    //   q = 0..3 : p_f[k]*num[k]      q = 4..7 : p_f[k]*den
    const float4 pf = s_lab[ty + RAD][tx + RAD];
    s_red[0][tid] = pf.x * num.x;
    s_red[1][tid] = pf.y * num.y;
    s_red[2][tid] = pf.z * num.z;
    s_red[3][tid] = pf.w * num.w;
    s_red[4][tid] = pf.x * den;
    s_red[5][tid] = pf.y * den;
    s_red[6][tid] = pf.z * den;
    s_red[7][tid] = pf.w * den;
    __syncthreads();

    // --- block reduction on the matrix pipe ---
    // Wave q reduces quantity q's 256 values. A = data (16x4 f32), B = ones:
    // D[m][n] += sum_k A[m][k], chained over 4 WMMAs => row-sums of all 256
    // values. Per-lane add of the 8 D VGPRs gives sum(rows 0..7) in lanes
    // 0..15 and sum(rows 8..15) in lanes 16..31; one shfl_xor(16) finishes.
    const int wave = tid >> 5;
    const int lane = tid & 31;

    float vbuf[8];
    #pragma unroll
    for (int j = 0; j < 8; ++j) vbuf[j] = s_red[wave][lane + 32 * j];

    const v2f one2 = {1.0f, 1.0f};
    v8f cacc = {};
    #pragma unroll
    for (int j = 0; j < 4; ++j) {
        v2f a = {vbuf[2 * j], vbuf[2 * j + 1]};
        cacc = __builtin_amdgcn_wmma_f32_16x16x4_f32(
            /*neg_a=*/false, a, /*neg_b=*/false, one2,
            /*c_mod=*/(short)0, cacc, /*reuse_a=*/false, /*reuse_b=*/false);
    }
    float s = cacc[0] + cacc[1] + cacc[2] + cacc[3] +
              cacc[4] + cacc[5] + cacc[6] + cacc[7];
    const float tot = s + __shfl_xor(s, 16, 32);   // wave32: pair halves

    if (lane == 0) atomicAdd(&acc[b * 8 + wave], tot);
}

// ---------------------------------------------------------------------------
// Stage 2: scalar combine  loss = K - (1/NB) * sum_{b,k} |numSum/denSum|
// ---------------------------------------------------------------------------
__global__ void ncut_stage2(const float* __restrict__ acc, float* __restrict__ out)
{
    if (threadIdx.x == 0 && blockIdx.x == 0) {
        float rs = 0.0f;
        for (int b = 0; b < NB; ++b)
            for (int k = 0; k < NK; ++k) {
                const float a = acc[b * 8 + k];
                const float d = acc[b * 8 + 4 + k];
                rs += fabsf(a / d);
            }
        out[0] = (float)NK - rs / (float)NB;
    }
}

// ---------------------------------------------------------------------------
extern "C" void kernel_launch(void* const* d_in, const int* in_sizes, int n_in,
                              void* d_out, int out_size, void* d_ws, size_t ws_size,
                              hipStream_t stream)
{
    const float* images = (const float*)d_in[0];   // [8,1,224,224] f32
    const float* labels = (const float*)d_in[1];   // [8,4,224,224] f32
    float* out = (float*)d_out;                    // scalar f32
    float* acc = (float*)d_ws;                     // 64 f32 accumulators

    ncut_zero<<<1, 64, 0, stream>>>(acc);

    dim3 grid(WWD / TILE, HH / TILE, NB);          // 14 x 14 x 8 tiles
    ncut_stage1<<<grid, 256, 0, stream>>>(images, labels, acc);

    ncut_stage2<<<1, 1, 0, stream>>>(acc, out);
}